// RelPosEmbedder_43044162241209
// MI455X (gfx1250) — compile-verified
//
#include <hip/hip_runtime.h>
#include <hip/hip_bf16.h>

typedef float v2f __attribute__((ext_vector_type(2)));
typedef float v8f __attribute__((ext_vector_type(8)));

#define CTOK      42
#define CZ        128
#define CREL      115
#define LSTRIDE   132          // 128 + 4 pad floats -> rotates LDS banks per row
#define ZROW      115          // first of 3 zero rows used for K-padding
#define NROWS     118
#define ENT_ROW   108          // entity_same column of W
#define CH_BASE   109          // rel_chain one-hot base row

__global__ __launch_bounds__(256, 1)
void relpos_embed_kernel(const float* __restrict__ rel_tok,
                         const float* __restrict__ W,
                         const int*   __restrict__ asym,
                         const int*   __restrict__ sym,
                         const int*   __restrict__ ent,
                         const int*   __restrict__ res,
                         float*       __restrict__ out,
                         int N, long long nTiles)
{
    __shared__ float Wl[NROWS * LSTRIDE];

    const int tid = threadIdx.x;

    // ---- stage W into LDS (bank-conflict-free stride), plus zero pad rows ----
    for (int idx = tid; idx < CREL * CZ; idx += 256) {
        int r = idx >> 7, c = idx & 127;
        Wl[r * LSTRIDE + c] = W[idx];
    }
    for (int idx = tid; idx < 3 * CZ; idx += 256) {
        int r = ZROW + (idx >> 7), c = idx & 127;
        Wl[r * LSTRIDE + c] = 0.0f;
    }
    __syncthreads();

    const int lane   = tid & 31;
    const int waveId = tid >> 5;
    const int hi     = lane >> 4;    // 0: lanes 0-15, 1: lanes 16-31
    const int lj     = lane & 15;

    const long long tileIdx = (long long)blockIdx.x * 8 + waveId;
    if (tileIdx >= nTiles) return;   // wave-uniform: EXEC stays all-ones past here
    const long long p0 = tileIdx * 16;          // 16 consecutive (i,j) pairs
    const int i  = (int)(p0 / N);               // constant over the tile (16 | N)
    const int j0 = (int)(p0 % N);
    const int j  = j0 + lj;                     // this lane's pair: m = lj

    // ---- per-lane structural features for its own row m = lj ----
    const bool chain_same  = (asym[i] == asym[j]);
    const bool entity_same = (ent[i]  == ent[j]);
    int off = res[i] - res[j] + 32;
    off = min(max(off, 0), 64);
    const int d_res  = chain_same ? off : 65;
    int co = sym[i] - sym[j] + 2;
    co = min(max(co, 0), 4);
    const int d_chain = (chain_same || !entity_same) ? 5 : co;
    const float ent_f = entity_same ? 1.0f : 0.0f;
    const int rowRes = d_res;
    const int rowCh  = CH_BASE + d_chain;

    // Broadcast per-row LUT data to match the C/D fragment layout:
    // acc VGPR r holds row m = r + 8*hi.
    int   resRow[8], chRow[8];
    float entR[8];
#pragma unroll
    for (int r = 0; r < 8; ++r) {
        const int src = r + (hi << 3);
        resRow[r] = __shfl(rowRes, src, 32);
        chRow[r]  = __shfl(rowCh,  src, 32);
        entR[r]   = __shfl(ent_f,  src, 32);
    }

    // ---- preload A fragments: 16x4 f32, K padded 42 -> 44 (11 steps) ----
    // Lane layout: a.x = K = 4s + 2*hi, a.y = K+1, row M = lj.
    const float* arow = rel_tok + (p0 + lj) * CTOK;
    v2f afrag[11];
#pragma unroll
    for (int s = 0; s < 11; ++s) {
        const int k  = 4 * s + 2 * hi;
        const int ks = min(k, 40);                         // safe aligned address
        v2f v = *(const v2f*)(arow + ks);                  // 8B-aligned: 42*4 & k even
        if (k >= CTOK) { v.x = 0.0f; v.y = 0.0f; }         // pad lanes -> zeros
        afrag[s] = v;
    }

    float* outp = out + p0 * CZ;

    // ---- 8 column tiles of 16 over C_Z = 128 ----
    for (int t = 0; t < 8; ++t) {
        const int n = t * 16 + lj;                         // this lane's column
        v8f acc = {0.f, 0.f, 0.f, 0.f, 0.f, 0.f, 0.f, 0.f};

#pragma unroll
        for (int s = 0; s < 11; ++s) {
            const int k  = 4 * s + 2 * hi;
            const int r0 = (k     < CTOK) ? (66 + k)     : ZROW;
            const int r1 = (k + 1 < CTOK) ? (66 + k + 1) : ZROW;
            v2f b;
            b.x = Wl[r0 * LSTRIDE + n];
            b.y = Wl[r1 * LSTRIDE + n];
            acc = __builtin_amdgcn_wmma_f32_16x16x4_f32(
                false, afrag[s], false, b, (short)0, acc, false, false);
        }

        // add one-hot / boolean contributions as row gathers from LDS, store NT
#pragma unroll
        for (int r = 0; r < 8; ++r) {
            const float lut = Wl[resRow[r] * LSTRIDE + n]
                            + Wl[chRow[r]  * LSTRIDE + n]
                            + entR[r] * Wl[ENT_ROW * LSTRIDE + n];
            const int m = r + 8 * hi;                      // C/D layout row
            __builtin_nontemporal_store(acc[r] + lut,
                                        outp + (long long)m * CZ + n);
        }
    }
}

extern "C" void kernel_launch(void* const* d_in, const int* in_sizes, int n_in,
                              void* d_out, int out_size, void* d_ws, size_t ws_size,
                              hipStream_t stream) {
    const float* rel_tok = (const float*)d_in[0];   // [1,N,N,42] f32
    const float* W       = (const float*)d_in[1];   // [115,128]  f32
    const int*   asym    = (const int*)d_in[2];     // [1,N]
    const int*   sym     = (const int*)d_in[3];     // [1,N]
    const int*   ent     = (const int*)d_in[4];     // [1,N]
    const int*   res     = (const int*)d_in[5];     // [1,N]
    float*       out     = (float*)d_out;           // [1,N,N,128] f32

    const int N = in_sizes[2];                      // 1024
    const long long pairs  = (long long)N * N;
    const long long nTiles = pairs / 16;            // N multiple of 16
    const int blocks = (int)((nTiles + 7) / 8);     // 8 waves per block

    relpos_embed_kernel<<<blocks, 256, 0, stream>>>(rel_tok, W, asym, sym, ent,
                                                    res, out, N, nTiles);
}